// MainGraph_62964220559445
// MI455X (gfx1250) — compile-verified
//
#include <hip/hip_runtime.h>
#include <hip/hip_bf16.h>
#include <math.h>

typedef __attribute__((ext_vector_type(2))) float v2f;
typedef __attribute__((ext_vector_type(8))) float v8f;

#define NN      50000
#define EE      800000
#define FD      128      // IN == H == K == 128 everywhere
#define HS      64       // S
#define NC      10       // C
#define NG      128      // G
#define BN_EPS  1e-5f
#define APAD    130      // padded LDS row stride (floats), even -> 8B aligned rows

// ---------------------------------------------------------------- utilities
__global__ void fill_kernel(float* __restrict__ p, size_t n, float v) {
    size_t i = (size_t)blockIdx.x * blockDim.x + threadIdx.x;
    size_t stride = (size_t)gridDim.x * blockDim.x;
    for (; i < n; i += stride) p[i] = v;
}

__device__ inline void atomicMaxF32(float* addr, float val) {
    unsigned int* ua = (unsigned int*)addr;
    unsigned int cur = *ua;
    while (__uint_as_float(cur) < val) {
        unsigned int assumed = cur;
        cur = atomicCAS(ua, assumed, __float_as_uint(val));
        if (cur == assumed) break;
    }
}

// --------------------------------------------------- BatchNorm: column stats
__global__ void bn_stats_kernel(const float* __restrict__ x, float* __restrict__ stats) {
    int f = threadIdx.x;
    int r0 = blockIdx.x * 128;
    int r1 = min(r0 + 128, NN);
    float s = 0.f, sq = 0.f;
    for (int r = r0; r < r1; ++r) {
        float v = x[(size_t)r * FD + f];
        s += v;
        sq += v * v;
    }
    atomicAdd(&stats[f], s);
    atomicAdd(&stats[FD + f], sq);
}

__global__ void bn_norm_kernel(const float* __restrict__ x,
                               const float* __restrict__ stats,
                               const float* __restrict__ gamma,
                               const float* __restrict__ beta,
                               float* __restrict__ h) {
    size_t idx = (size_t)blockIdx.x * blockDim.x + threadIdx.x;
    if (idx >= (size_t)NN * FD) return;
    int f = (int)(idx & (FD - 1));
    float mu  = stats[f] * (1.0f / NN);
    float var = stats[FD + f] * (1.0f / NN) - mu * mu;
    float scl = rsqrtf(var + BN_EPS) * gamma[f];
    h[idx] = (x[idx] - mu) * scl + beta[f];
}

// ------------------------------------------------------------- graph kernels
__global__ void degree_kernel(const int* __restrict__ ei, float* __restrict__ deg) {
    int e = blockIdx.x * blockDim.x + threadIdx.x;
    if (e >= EE) return;
    atomicAdd(&deg[ei[EE + e]], 1.0f);
}

// 32 lanes per edge, 4 features per lane (float4 gather + 4 f32 atomic adds)
__global__ void scatter_kernel(const float* __restrict__ h,
                               const int* __restrict__ ei,
                               float* __restrict__ agg) {
    size_t t = (size_t)blockIdx.x * blockDim.x + threadIdx.x;
    if (t >= (size_t)EE * 32) return;
    int e    = (int)(t >> 5);
    int lane = (int)(t & 31);
    int f = lane * 4;
    int src = ei[e];
    int dst = ei[EE + e];
    const float4 v = *(const float4*)(h + (size_t)src * FD + f);
    float* a = agg + (size_t)dst * FD + f;
    atomicAdd(a + 0, v.x);
    atomicAdd(a + 1, v.y);
    atomicAdd(a + 2, v.z);
    atomicAdd(a + 3, v.w);
}

__global__ void divdeg_kernel(float* __restrict__ agg, const float* __restrict__ deg) {
    size_t idx = (size_t)blockIdx.x * blockDim.x + threadIdx.x;
    if (idx >= (size_t)NN * FD) return;
    float d = fmaxf(deg[idx >> 7], 1.0f);
    agg[idx] = agg[idx] / d;
}

// --------------------------------------------------------- WMMA f32 GEMM
// Block = 256 threads (8 waves), one block per 16-row panel of:
//   out = A1 @ W1^T (+ A2 @ W2^T) + bias, optional ELU,  K fixed = 128.
// A-panel rows are staged in LDS once (padded stride APAD), shared by all
// column-tile waves. Wave w computes output columns [16w, 16w+16).
// Fragment layout per CDNA5 ISA (16x4 f32):
//   lane<16 : {K=k,K=k+1} of row(lane),  lane>=16: {K=k+2,K=k+3}
// C/D: lane<16 -> rows tileM+0..7, col tileN+lane; lane>=16 -> rows +8..15.
__global__ void wmma_gemm_kernel(const float* __restrict__ A1,
                                 const float* __restrict__ W1,
                                 const float* __restrict__ A2,
                                 const float* __restrict__ W2,
                                 const float* __restrict__ bias,
                                 float* __restrict__ out,
                                 int Nout, int ldOut, int apply_elu) {
    __shared__ float sA[2 * 16 * APAD];

    const int tileM = blockIdx.x * 16;
    const int tid   = threadIdx.x;

    // ---- cooperative stage of A panels into LDS (coalesced: tid -> k)
    for (int idx = tid; idx < 16 * FD; idx += 256) {
        int r = idx >> 7;
        int k = idx & (FD - 1);
        sA[r * APAD + k] = A1[(size_t)(tileM + r) * FD + k];
    }
    if (A2) {
        float* s2 = sA + 16 * APAD;
        for (int idx = tid; idx < 16 * FD; idx += 256) {
            int r = idx >> 7;
            int k = idx & (FD - 1);
            s2[r * APAD + k] = A2[(size_t)(tileM + r) * FD + k];
        }
    }
    __syncthreads();

    const int wave  = tid >> 5;
    const int tileN = wave * 16;
    if (tileN >= Nout) return;          // wave-uniform; active waves keep EXEC=all-1

    const int lane = tid & 31;
    const int half = lane >> 4;
    const int r    = lane & 15;

    v8f c = {0.f, 0.f, 0.f, 0.f, 0.f, 0.f, 0.f, 0.f};

    const float* sA1r = sA + r * APAD;
    const float* sA2r = sA + 16 * APAD + r * APAD;
    const size_t wrow = (size_t)(tileN + r) * FD;

#pragma unroll 8
    for (int k = 0; k < FD; k += 4) {
        const int kc = k + 2 * half;
        v2f a = *(const v2f*)(sA1r + kc);                // ds_load_b64
        v2f b = *(const v2f*)(W1 + wrow + kc);           // global_load_b64 (L2-hot)
        c = __builtin_amdgcn_wmma_f32_16x16x4_f32(false, a, false, b,
                                                  (short)0, c, false, false);
    }
    if (A2) {
#pragma unroll 8
        for (int k = 0; k < FD; k += 4) {
            const int kc = k + 2 * half;
            v2f a = *(const v2f*)(sA2r + kc);
            v2f b = *(const v2f*)(W2 + wrow + kc);
            c = __builtin_amdgcn_wmma_f32_16x16x4_f32(false, a, false, b,
                                                      (short)0, c, false, false);
        }
    }

    const int col  = tileN + r;
    const int row0 = tileM + half * 8;
    const float bv = bias ? bias[col] : 0.f;
#pragma unroll
    for (int i = 0; i < 8; ++i) {
        float v = c[i] + bv;
        if (apply_elu) v = (v > 0.f) ? v : (expf(v) - 1.f);
        out[(size_t)(row0 + i) * ldOut + col] = v;
    }
}

// ------------------------------------------------------------- max pool
__global__ void maxpool_kernel(const float* __restrict__ h,
                               const int* __restrict__ batch,
                               float* __restrict__ readout) {
    size_t idx = (size_t)blockIdx.x * blockDim.x + threadIdx.x;
    if (idx >= (size_t)NN * FD) return;
    int n = (int)(idx >> 7);
    int f = (int)(idx & (FD - 1));
    int g = batch[n];
    atomicMaxF32(&readout[(size_t)g * FD + f], h[idx]);
}

// --------------------------------------------------- final tiny dense [128,10]
__global__ void final_kernel(const float* __restrict__ rv,
                             const float* __restrict__ Wc2,
                             const float* __restrict__ bc2,
                             float* __restrict__ out) {
    int idx = blockIdx.x * blockDim.x + threadIdx.x;
    if (idx >= NG * NC) return;
    int g = idx / NC;
    int cc = idx % NC;
    float acc = bc2[cc];
    const float* rvg = rv + (size_t)g * HS;
    const float* w   = Wc2 + (size_t)cc * HS;
#pragma unroll 4
    for (int s = 0; s < HS; ++s) acc += rvg[s] * w[s];
    out[idx] = acc;
}

// ---------------------------------------------------------------- launcher
extern "C" void kernel_launch(void* const* d_in, const int* in_sizes, int n_in,
                              void* d_out, int out_size, void* d_ws, size_t ws_size,
                              hipStream_t stream) {
    const float* x     = (const float*)d_in[0];
    const int*   ei    = (const int*)d_in[1];
    const int*   batch = (const int*)d_in[2];
    const float* gamma = (const float*)d_in[3];
    const float* beta  = (const float*)d_in[4];
    const float* W1l   = (const float*)d_in[5];
    const float* b1    = (const float*)d_in[6];
    const float* W1r   = (const float*)d_in[7];
    const float* W2l   = (const float*)d_in[8];
    const float* b2    = (const float*)d_in[9];
    const float* W2r   = (const float*)d_in[10];
    const float* Wc1   = (const float*)d_in[11];
    const float* bc1   = (const float*)d_in[12];
    const float* Wc2   = (const float*)d_in[13];
    const float* bc2   = (const float*)d_in[14];

    float* out = (float*)d_out;            // [128,10]
    float* rv  = out + NG * NC;            // [128,64]

    // workspace layout (floats)
    float* ws      = (float*)d_ws;
    float* h0      = ws;                               // [N,128]  BN out / layer2 out
    float* h1      = h0 + (size_t)NN * FD;             // [N,128]  layer1 out
    float* agg     = h1 + (size_t)NN * FD;             // [N,128]  aggregation buffer
    float* deg     = agg + (size_t)NN * FD;            // [N]
    float* stats   = deg + NN;                         // [256]
    float* readout = stats + 256;                      // [128,128]

    const size_t NF = (size_t)NN * FD;
    const int TB = 256;
    const int gNF = (int)((NF + TB - 1) / TB);

    // ---- BatchNorm
    fill_kernel<<<1, TB, 0, stream>>>(stats, 256, 0.f);
    bn_stats_kernel<<<(NN + 127) / 128, 128, 0, stream>>>(x, stats);
    bn_norm_kernel<<<gNF, TB, 0, stream>>>(x, stats, gamma, beta, h0);

    // ---- degree (shared by both layers)
    fill_kernel<<<(NN + TB - 1) / TB, TB, 0, stream>>>(deg, NN, 0.f);
    degree_kernel<<<(EE + TB - 1) / TB, TB, 0, stream>>>(ei, deg);

    // ---- SAGE layer 1
    fill_kernel<<<gNF, TB, 0, stream>>>(agg, NF, 0.f);
    {
        size_t total = (size_t)EE * 32;
        scatter_kernel<<<(int)((total + TB - 1) / TB), TB, 0, stream>>>(h0, ei, agg);
    }
    divdeg_kernel<<<gNF, TB, 0, stream>>>(agg, deg);
    wmma_gemm_kernel<<<NN / 16, 256, 0, stream>>>(agg, W1l, h0, W1r, b1, h1,
                                                  FD, FD, /*elu=*/1);

    // ---- SAGE layer 2
    fill_kernel<<<gNF, TB, 0, stream>>>(agg, NF, 0.f);
    {
        size_t total = (size_t)EE * 32;
        scatter_kernel<<<(int)((total + TB - 1) / TB), TB, 0, stream>>>(h1, ei, agg);
    }
    divdeg_kernel<<<gNF, TB, 0, stream>>>(agg, deg);
    wmma_gemm_kernel<<<NN / 16, 256, 0, stream>>>(agg, W2l, h1, W2r, b2, h0,
                                                  FD, FD, /*elu=*/1);

    // ---- global max pool
    fill_kernel<<<(NG * FD + TB - 1) / TB, TB, 0, stream>>>(readout, (size_t)NG * FD, -INFINITY);
    maxpool_kernel<<<gNF, TB, 0, stream>>>(h0, batch, readout);

    // ---- classify1 : rv = readout @ Wc1^T + bc1   (M=128, K=128, Nout=64)
    wmma_gemm_kernel<<<NG / 16, 256, 0, stream>>>(readout, Wc1, nullptr, nullptr,
                                                  bc1, rv, HS, HS, /*elu=*/0);

    // ---- classify2 : out = rv @ Wc2^T + bc2   (tiny, scalar)
    final_kernel<<<(NG * NC + TB - 1) / TB, TB, 0, stream>>>(rv, Wc2, bc2, out);
}